// StructuralLearning_89077621719710
// MI455X (gfx1250) — compile-verified
//
#include <hip/hip_runtime.h>
#include <hip/hip_bf16.h>

// CDNA5 / gfx1250: wave32, WMMA f32 16x16x4.
typedef __attribute__((ext_vector_type(2))) float v2f;
typedef __attribute__((ext_vector_type(8))) float v8f;

#define B_SZ 64
#define H_SZ 128
#define D_SZ 768
#define TM   4          // M-tiles (16 rows each) per wave => 64 rows/wave

// ---------------------------------------------------------------------------
// fp32 WMMA GEMM:  C[M,N] = act(A[M,K] @ W[K,N] + bias[N])
// One wave32 computes a 64x16 output slab (4 accumulators) so each B fragment
// (2 strided b32 loads) feeds FOUR v_wmma_f32_16x16x4_f32 ops; the 4
// independent WMMAs also hide XDL + load latency.
//
// ISA 32-bit fragment layouts (CDNA5 ISA §7.12.2):
//   A 16x4 : lanes 0-15 -> M=lane, v0=K0,v1=K1 ; lanes 16-31 -> M=lane-16, v0=K2,v1=K3
//   B 4x16 : lanes 0-15 -> N=lane, v0=K0,v1=K1 ; lanes 16-31 -> N=lane-16, v0=K2,v1=K3
//   C 16x16: lanes 0-15 -> N=lane, M=vgpr(0..7); lanes 16-31 -> N=lane-16, M=vgpr+8
// ---------------------------------------------------------------------------
__global__ void wmma_gemm_bias_act(const float* __restrict__ A,
                                   const float* __restrict__ W,
                                   const float* __restrict__ bias,
                                   float* __restrict__ C,
                                   int M, int K, int N, int do_relu) {
    const int lane = threadIdx.x & 31;        // wave32
    const int half = lane >> 4;               // 0: K0/K1, 1: K2/K3
    const int l    = lane & 15;

    const int tile_n = blockIdx.x * 16;
    const int tile_m = blockIdx.y * (16 * TM);

    const float* __restrict__ Arow[TM];
    #pragma unroll
    for (int t = 0; t < TM; ++t)
        Arow[t] = A + (size_t)(tile_m + t * 16 + l) * K;
    const float* __restrict__ Wcol = W + (size_t)(tile_n + l);

    v8f acc[TM];
    #pragma unroll
    for (int t = 0; t < TM; ++t) acc[t] = (v8f){};

    #pragma unroll 2
    for (int k = 0; k < K; k += 4) {
        const int kk = k + 2 * half;

        // B fragment: shared by all 4 M-tiles this step.
        v2f bfrag;
        bfrag.x = Wcol[(size_t)kk * N];
        bfrag.y = Wcol[(size_t)(kk + 1) * N];

        // A fragments: contiguous pairs -> global_load_b64 each.
        v2f afrag[TM];
        #pragma unroll
        for (int t = 0; t < TM; ++t) {
            afrag[t] = *(const v2f*)(Arow[t] + kk);
            __builtin_prefetch(Arow[t] + kk + 64, 0, 1);   // global_prefetch_b8
        }

        // 4 independent D = A x B + C  (v_wmma_f32_16x16x4_f32)
        #pragma unroll
        for (int t = 0; t < TM; ++t)
            acc[t] = __builtin_amdgcn_wmma_f32_16x16x4_f32(
                /*neg_a=*/false, afrag[t],
                /*neg_b=*/false, bfrag,
                /*c_mod=*/(short)0, acc[t],
                /*reuse_a=*/false, /*reuse_b=*/false);
    }

    const int col = tile_n + l;
    const float bv = bias ? bias[col] : 0.0f;

    #pragma unroll
    for (int t = 0; t < TM; ++t) {
        const int row0 = tile_m + t * 16 + half * 8;
        #pragma unroll
        for (int r = 0; r < 8; ++r) {
            float v = acc[t][r] + bv;
            if (do_relu) v = fmaxf(v, 0.0f);
            C[(size_t)(row0 + r) * N + col] = v;
        }
    }
}

// ---------------------------------------------------------------------------
// Fused pairwise stage:
//   adj[b,i,j] = sigmoid( sum_k relu(a[b,i,k] + bb[b,j,k]) * S2[k] + s2b )
// (s1b already folded into `a` by the preceding GEMM's bias epilogue.)
// Block = 256 threads computes a 16x16 (i,j) tile for one batch. The relu
// inside the contraction forbids a GEMM formulation, so this stays VALU, but
// nothing of the [B,H,H,128] intermediate is ever materialized.
// LDS row pitch 129 floats => (row + k) mod 64 banks: 16 distinct banks for
// the 16-way fan-out across i (resp. j), no serialization.
// ---------------------------------------------------------------------------
__global__ void pairwise_adj(const float* __restrict__ a,
                             const float* __restrict__ bb,
                             const float* __restrict__ S2,
                             const float* __restrict__ s2b,
                             float* __restrict__ adj) {
    __shared__ float a_t[16][129];
    __shared__ float b_t[16][129];
    __shared__ float s2s[H_SZ];

    const int tid = threadIdx.x;              // 0..255
    const int i0  = blockIdx.x * 16;
    const int j0  = blockIdx.y * 16;
    const int bi  = blockIdx.z;               // batch index

    for (int idx = tid; idx < 16 * H_SZ; idx += 256) {
        const int r = idx >> 7;               // 0..15
        const int c = idx & (H_SZ - 1);       // 0..127
        a_t[r][c] = a[((size_t)(bi * H_SZ + i0 + r)) * H_SZ + c];
        b_t[r][c] = bb[((size_t)(bi * H_SZ + j0 + r)) * H_SZ + c];
    }
    if (tid < H_SZ) s2s[tid] = S2[tid];
    __syncthreads();

    const int i = tid >> 4;                   // 0..15
    const int j = tid & 15;                   // 0..15

    float sum = s2b[0];
    #pragma unroll 8
    for (int k = 0; k < H_SZ; ++k) {
        const float t = a_t[i][k] + b_t[j][k];
        sum = fmaf(fmaxf(t, 0.0f), s2s[k], sum);
    }
    const float sig = 1.0f / (1.0f + __expf(-sum));
    adj[((size_t)(bi * H_SZ + i0 + i)) * H_SZ + (j0 + j)] = sig;
}

// ---------------------------------------------------------------------------
// Launch: 6 WMMA GEMMs + 1 fused pairwise kernel.
// Inputs (dict order): 0 image, 1 text, 2 W1, 3 b1, 4 W2, 5 b2,
//                      6 S1[2H,H], 7 s1b, 8 S2[H,1], 9 s2b.
// Workspace (16 MB): [t1_img | t1_txt | img_h | txt_h]; `a` reuses t1_img,
// `b_` reuses t1_txt once the W2 GEMMs have consumed them.
// ---------------------------------------------------------------------------
extern "C" void kernel_launch(void* const* d_in, const int* in_sizes, int n_in,
                              void* d_out, int out_size, void* d_ws, size_t ws_size,
                              hipStream_t stream) {
    const float* img = (const float*)d_in[0];
    const float* txt = (const float*)d_in[1];
    const float* W1  = (const float*)d_in[2];
    const float* b1  = (const float*)d_in[3];
    const float* W2  = (const float*)d_in[4];
    const float* b2  = (const float*)d_in[5];
    const float* S1  = (const float*)d_in[6];
    const float* s1b = (const float*)d_in[7];
    const float* S2  = (const float*)d_in[8];
    const float* s2b = (const float*)d_in[9];
    float* out = (float*)d_out;

    const int M = B_SZ * H_SZ;                          // 8192 rows
    const size_t bufElems = (size_t)M * H_SZ;           // 1M floats = 4 MB
    float* t1i = (float*)d_ws;                          // relu layer, image
    float* t1t = t1i + bufElems;                        // relu layer, text
    float* hi  = t1t + bufElems;                        // img_h
    float* ht  = hi  + bufElems;                        // txt_h
    float* av  = t1i;                                   // a  (reuses t1i)
    float* bv  = t1t;                                   // b_ (reuses t1t)

    const dim3 wave(32, 1, 1);
    const dim3 g1(H_SZ / 16, M / (16 * TM));            // 8 x 128 waves

    // Layer 1: relu(X @ W1 + b1)   [8192x768]x[768x128]
    wmma_gemm_bias_act<<<g1, wave, 0, stream>>>(img, W1, b1, t1i, M, D_SZ, H_SZ, 1);
    wmma_gemm_bias_act<<<g1, wave, 0, stream>>>(txt, W1, b1, t1t, M, D_SZ, H_SZ, 1);

    // Layer 2: t1 @ W2 + b2        [8192x128]x[128x128]
    wmma_gemm_bias_act<<<g1, wave, 0, stream>>>(t1i, W2, b2, hi, M, H_SZ, H_SZ, 0);
    wmma_gemm_bias_act<<<g1, wave, 0, stream>>>(t1t, W2, b2, ht, M, H_SZ, H_SZ, 0);

    // a = img_h @ S1[:H] + s1b ; b_ = txt_h @ S1[H:]
    wmma_gemm_bias_act<<<g1, wave, 0, stream>>>(hi, S1, s1b, av, M, H_SZ, H_SZ, 0);
    wmma_gemm_bias_act<<<g1, wave, 0, stream>>>(ht, S1 + (size_t)H_SZ * H_SZ, nullptr,
                                                bv, M, H_SZ, H_SZ, 0);

    // adj[b,i,j] = sigmoid(relu(a_i + b_j) . S2 + s2b), fused
    const dim3 gp(H_SZ / 16, H_SZ / 16, B_SZ);
    pairwise_adj<<<gp, dim3(256, 1, 1), 0, stream>>>(av, bv, S2, s2b, out);
}